// DiscriminativeLoss_80453327389032
// MI455X (gfx1250) — compile-verified
//
#include <hip/hip_runtime.h>
#include <math.h>

typedef __attribute__((ext_vector_type(2))) float v2f;
typedef __attribute__((ext_vector_type(8))) float v8f;

#define EPSV 1e-8f
#define NBLK 256          // blocks in streaming passes
#define TPB  256          // 8 waves/block
#define NWAVES (NBLK * 8) // 2048 waves total
#define P1_STRIDE 1088    // 64*16 sums + 64 counts

// One unguarded 4-point WMMA step for pass 1 (segment sums + counts).
// p0 = group base + 2*kh (even), lanes supply A one-hot + B feature elems.
__device__ __forceinline__ void p1_step(const float* __restrict__ feat,
                                        const int* __restrict__ lab,
                                        long p0, int n,
                                        v8f& a0, v8f& a1, v8f& a2, v8f& a3,
                                        float& c0, float& c1, float& c2, float& c3) {
  long long lv = __builtin_nontemporal_load((const long long*)(lab + p0));
  int l0 = (int)(unsigned)(lv & 0xffffffffLL);
  int l1 = (int)(unsigned)((unsigned long long)lv >> 32);
  v2f b;
  b.x = __builtin_nontemporal_load(feat + p0 * 16 + n);
  b.y = __builtin_nontemporal_load(feat + p0 * 16 + 16 + n);

  v2f a;
  a.x = (l0 == n)      ? 1.f : 0.f;
  a.y = (l1 == n)      ? 1.f : 0.f;
  c0 += a.x + a.y;
  a0 = __builtin_amdgcn_wmma_f32_16x16x4_f32(false, a, false, b, (short)0, a0, false, false);
  a.x = (l0 == n + 16) ? 1.f : 0.f;
  a.y = (l1 == n + 16) ? 1.f : 0.f;
  c1 += a.x + a.y;
  a1 = __builtin_amdgcn_wmma_f32_16x16x4_f32(false, a, false, b, (short)0, a1, false, false);
  a.x = (l0 == n + 32) ? 1.f : 0.f;
  a.y = (l1 == n + 32) ? 1.f : 0.f;
  c2 += a.x + a.y;
  a2 = __builtin_amdgcn_wmma_f32_16x16x4_f32(false, a, false, b, (short)0, a2, false, false);
  a.x = (l0 == n + 48) ? 1.f : 0.f;
  a.y = (l1 == n + 48) ? 1.f : 0.f;
  c3 += a.x + a.y;
  a3 = __builtin_amdgcn_wmma_f32_16x16x4_f32(false, a, false, b, (short)0, a3, false, false);
}

// ---------------------------------------------------------------------------
// Pass 1: segment sums + counts via V_WMMA_F32_16X16X4_F32.
// ---------------------------------------------------------------------------
__global__ __launch_bounds__(TPB) void dl_pass1(const float* __restrict__ feat,
                                                const int* __restrict__ lab,
                                                float* __restrict__ P1, int N) {
  __shared__ float lsum[8 * 1024];  // per-wave 64x16 sums
  __shared__ float lcnt[8 * 128];   // per-wave, per-half 64 counts

  const int tid  = threadIdx.x;
  const int wave = tid >> 5;
  const int lane = tid & 31;
  const int kh   = lane >> 4;   // low lanes: K=0,1 ; high lanes: K=2,3
  const int n    = lane & 15;   // column N == one-hot row M
  const long W   = (long)blockIdx.x * 8 + wave;
  const long stride = (long)NWAVES * 4;

  v8f a0 = {}, a1 = {}, a2 = {}, a3 = {};
  float c0 = 0.f, c1 = 0.f, c2 = 0.f, c3 = 0.f;

  // Branch-free main loop: base multiple of 4, base+4 <= N => whole group valid.
  long base = W * 4;
  for (; base + stride + 4 <= N; base += 2 * stride) {
    p1_step(feat, lab, base + 2 * kh,          n, a0, a1, a2, a3, c0, c1, c2, c3);
    p1_step(feat, lab, base + stride + 2 * kh, n, a0, a1, a2, a3, c0, c1, c2, c3);
  }
  for (; base + 4 <= N; base += stride) {
    p1_step(feat, lab, base + 2 * kh, n, a0, a1, a2, a3, c0, c1, c2, c3);
  }
  // Tail (<=3 points): one designated wave, guarded.
  if (W == 0 && (N & 3)) {
    long tb = (long)N & ~3L;
    long p0 = tb + 2 * kh, p1 = p0 + 1;
    int l0 = (p0 < N) ? lab[p0] : -1;
    int l1 = (p1 < N) ? lab[p1] : -1;
    v2f b;
    b.x = (p0 < N) ? feat[p0 * 16 + n] : 0.f;
    b.y = (p1 < N) ? feat[p1 * 16 + n] : 0.f;
    v2f a;
    a.x = (l0 == n)      ? 1.f : 0.f;
    a.y = (l1 == n)      ? 1.f : 0.f;
    c0 += a.x + a.y;
    a0 = __builtin_amdgcn_wmma_f32_16x16x4_f32(false, a, false, b, (short)0, a0, false, false);
    a.x = (l0 == n + 16) ? 1.f : 0.f;
    a.y = (l1 == n + 16) ? 1.f : 0.f;
    c1 += a.x + a.y;
    a1 = __builtin_amdgcn_wmma_f32_16x16x4_f32(false, a, false, b, (short)0, a1, false, false);
    a.x = (l0 == n + 32) ? 1.f : 0.f;
    a.y = (l1 == n + 32) ? 1.f : 0.f;
    c2 += a.x + a.y;
    a2 = __builtin_amdgcn_wmma_f32_16x16x4_f32(false, a, false, b, (short)0, a2, false, false);
    a.x = (l0 == n + 48) ? 1.f : 0.f;
    a.y = (l1 == n + 48) ? 1.f : 0.f;
    c3 += a.x + a.y;
    a3 = __builtin_amdgcn_wmma_f32_16x16x4_f32(false, a, false, b, (short)0, a3, false, false);
  }

  // Dump accumulators: C/D layout -> element (m = r + 8*kh, col = n)
#pragma unroll
  for (int r = 0; r < 8; ++r) {
    int m = r + 8 * kh;
    lsum[wave * 1024 + (0 * 16 + m) * 16 + n] = a0[r];
    lsum[wave * 1024 + (1 * 16 + m) * 16 + n] = a1[r];
    lsum[wave * 1024 + (2 * 16 + m) * 16 + n] = a2[r];
    lsum[wave * 1024 + (3 * 16 + m) * 16 + n] = a3[r];
  }
  lcnt[wave * 128 + kh * 64 + 0 * 16 + n] = c0;
  lcnt[wave * 128 + kh * 64 + 1 * 16 + n] = c1;
  lcnt[wave * 128 + kh * 64 + 2 * 16 + n] = c2;
  lcnt[wave * 128 + kh * 64 + 3 * 16 + n] = c3;
  __syncthreads();

  // Deterministic per-block reduce of 8 waves -> block partial
  for (int idx = tid; idx < P1_STRIDE; idx += TPB) {
    float s = 0.f;
    if (idx < 1024) {
#pragma unroll
      for (int w = 0; w < 8; ++w) s += lsum[w * 1024 + idx];
    } else {
      int c = idx - 1024;
#pragma unroll
      for (int w = 0; w < 8; ++w) s += lcnt[w * 128 + c] + lcnt[w * 128 + 64 + c];
    }
    P1[(long)blockIdx.x * P1_STRIDE + idx] = s;
  }
}

// Reduce block partials in fixed order -> S[1088] (1024 sums + 64 counts)
__global__ void dl_reduce1(const float* __restrict__ P1, float* __restrict__ S) {
  int idx = blockIdx.x * 256 + threadIdx.x;
  if (idx >= P1_STRIDE) return;
  float s = 0.f;
  for (int b = 0; b < NBLK; ++b) s += P1[(long)b * P1_STRIDE + idx];
  S[idx] = s;
}

// Single block: means, pairwise dist loss, reg loss
__global__ __launch_bounds__(256) void dl_means_pairs(const float* __restrict__ S,
                                                      float* __restrict__ means,
                                                      float* __restrict__ scal) {
  __shared__ float lm[1024];
  __shared__ float redA[256];
  __shared__ float redB[256];
  const int tid = threadIdx.x;

  for (int idx = tid; idx < 1024; idx += 256) {
    int c = idx >> 4;
    float m = S[idx] / fmaxf(S[1024 + c], 1.f);
    lm[idx] = m;
    means[idx] = m;
  }
  __syncthreads();

  float pd = 0.f;
  for (int p = tid; p < 4096; p += 256) {
    int i = p >> 6, j = p & 63;
    if (i != j) {
      float d2 = 0.f;
#pragma unroll
      for (int d = 0; d < 16; ++d) {
        float df = lm[i * 16 + d] - lm[j * 16 + d] + EPSV;
        d2 += df * df;
      }
      float h = fmaxf(3.0f - sqrtf(d2), 0.f);  // 2*DELTA_DIST = 3.0
      pd += h * h;
    }
  }
  float pr = 0.f;
  if (tid < 64) {
    float d2 = 0.f;
#pragma unroll
    for (int d = 0; d < 16; ++d) {
      float v = lm[tid * 16 + d] + EPSV;
      d2 += v * v;
    }
    pr = sqrtf(d2);
  }
  redA[tid] = pd;
  redB[tid] = pr;
  __syncthreads();
  for (int s = 128; s > 0; s >>= 1) {
    if (tid < s) { redA[tid] += redA[tid + s]; redB[tid] += redB[tid + s]; }
    __syncthreads();
  }
  if (tid == 0) {
    scal[0] = redA[0] / 4032.f;  // C*(C-1)
    scal[1] = redB[0] / 64.f;
  }
}

// One unguarded 4-point WMMA step for pass 2 (hinge^2 segment sum).
__device__ __forceinline__ void p2_step(const float* __restrict__ feat,
                                        const int* __restrict__ lab,
                                        const float* __restrict__ means,
                                        long p0, int n,
                                        v8f& a0, v8f& a1, v8f& a2, v8f& a3) {
  long long lv = __builtin_nontemporal_load((const long long*)(lab + p0));
  int l0 = (int)(unsigned)(lv & 0xffffffffLL);
  int l1 = (int)(unsigned)((unsigned long long)lv >> 32);
  float f0 = __builtin_nontemporal_load(feat + p0 * 16 + n);
  float f1 = __builtin_nontemporal_load(feat + p0 * 16 + 16 + n);
  float d0 = f0 - means[l0 * 16 + n] + EPSV;
  float d1 = f1 - means[l1 * 16 + n] + EPSV;
  float s0 = d0 * d0;
  float s1 = d1 * d1;
#pragma unroll
  for (int msk = 1; msk <= 8; msk <<= 1) {  // reduce 16 dims within half
    s0 += __shfl_xor(s0, msk, 32);
    s1 += __shfl_xor(s1, msk, 32);
  }
  float h0 = fmaxf(sqrtf(s0) - 0.5f, 0.f);  // DELTA_VAR
  float h1 = fmaxf(sqrtf(s1) - 0.5f, 0.f);

  v2f b;
  b.x = (n == 0) ? h0 * h0 : 0.f;  // B column 0 only
  b.y = (n == 0) ? h1 * h1 : 0.f;

  v2f a;
  a.x = (l0 == n)      ? 1.f : 0.f;
  a.y = (l1 == n)      ? 1.f : 0.f;
  a0 = __builtin_amdgcn_wmma_f32_16x16x4_f32(false, a, false, b, (short)0, a0, false, false);
  a.x = (l0 == n + 16) ? 1.f : 0.f;
  a.y = (l1 == n + 16) ? 1.f : 0.f;
  a1 = __builtin_amdgcn_wmma_f32_16x16x4_f32(false, a, false, b, (short)0, a1, false, false);
  a.x = (l0 == n + 32) ? 1.f : 0.f;
  a.y = (l1 == n + 32) ? 1.f : 0.f;
  a2 = __builtin_amdgcn_wmma_f32_16x16x4_f32(false, a, false, b, (short)0, a2, false, false);
  a.x = (l0 == n + 48) ? 1.f : 0.f;
  a.y = (l1 == n + 48) ? 1.f : 0.f;
  a3 = __builtin_amdgcn_wmma_f32_16x16x4_f32(false, a, false, b, (short)0, a3, false, false);
}

// ---------------------------------------------------------------------------
// Pass 2: per-point hinge^2, segment-summed per cluster via one-hot WMMA.
// ---------------------------------------------------------------------------
__global__ __launch_bounds__(TPB) void dl_pass2(const float* __restrict__ feat,
                                                const int* __restrict__ lab,
                                                const float* __restrict__ means,
                                                float* __restrict__ P2, int N) {
  __shared__ float lv[8 * 64];

  const int tid  = threadIdx.x;
  const int wave = tid >> 5;
  const int lane = tid & 31;
  const int kh   = lane >> 4;
  const int n    = lane & 15;
  const long W   = (long)blockIdx.x * 8 + wave;
  const long stride = (long)NWAVES * 4;

  v8f a0 = {}, a1 = {}, a2 = {}, a3 = {};

  long base = W * 4;
  for (; base + stride + 4 <= N; base += 2 * stride) {
    p2_step(feat, lab, means, base + 2 * kh,          n, a0, a1, a2, a3);
    p2_step(feat, lab, means, base + stride + 2 * kh, n, a0, a1, a2, a3);
  }
  for (; base + 4 <= N; base += stride) {
    p2_step(feat, lab, means, base + 2 * kh, n, a0, a1, a2, a3);
  }
  if (W == 0 && (N & 3)) {
    long tb = (long)N & ~3L;
    long p0 = tb + 2 * kh, p1 = p0 + 1;
    int l0 = (p0 < N) ? lab[p0] : -1;
    int l1 = (p1 < N) ? lab[p1] : -1;
    float f0 = (p0 < N) ? feat[p0 * 16 + n] : 0.f;
    float f1 = (p1 < N) ? feat[p1 * 16 + n] : 0.f;
    float m0 = (l0 >= 0) ? means[l0 * 16 + n] : 0.f;
    float m1 = (l1 >= 0) ? means[l1 * 16 + n] : 0.f;
    float d0 = f0 - m0 + EPSV;
    float d1 = f1 - m1 + EPSV;
    float s0 = d0 * d0, s1 = d1 * d1;
#pragma unroll
    for (int msk = 1; msk <= 8; msk <<= 1) {
      s0 += __shfl_xor(s0, msk, 32);
      s1 += __shfl_xor(s1, msk, 32);
    }
    float h0 = fmaxf(sqrtf(s0) - 0.5f, 0.f);
    float h1 = fmaxf(sqrtf(s1) - 0.5f, 0.f);
    float h2_0 = (p0 < N) ? h0 * h0 : 0.f;
    float h2_1 = (p1 < N) ? h1 * h1 : 0.f;
    v2f b;
    b.x = (n == 0) ? h2_0 : 0.f;
    b.y = (n == 0) ? h2_1 : 0.f;
    v2f a;
    a.x = (l0 == n)      ? 1.f : 0.f;
    a.y = (l1 == n)      ? 1.f : 0.f;
    a0 = __builtin_amdgcn_wmma_f32_16x16x4_f32(false, a, false, b, (short)0, a0, false, false);
    a.x = (l0 == n + 16) ? 1.f : 0.f;
    a.y = (l1 == n + 16) ? 1.f : 0.f;
    a1 = __builtin_amdgcn_wmma_f32_16x16x4_f32(false, a, false, b, (short)0, a1, false, false);
    a.x = (l0 == n + 32) ? 1.f : 0.f;
    a.y = (l1 == n + 32) ? 1.f : 0.f;
    a2 = __builtin_amdgcn_wmma_f32_16x16x4_f32(false, a, false, b, (short)0, a2, false, false);
    a.x = (l0 == n + 48) ? 1.f : 0.f;
    a.y = (l1 == n + 48) ? 1.f : 0.f;
    a3 = __builtin_amdgcn_wmma_f32_16x16x4_f32(false, a, false, b, (short)0, a3, false, false);
  }

  // Column 0 lives in lanes with n==0 (lane 0: M=r, lane 16: M=r+8)
  if (n == 0) {
#pragma unroll
    for (int r = 0; r < 8; ++r) {
      int m = r + 8 * kh;
      lv[wave * 64 + 0 * 16 + m] = a0[r];
      lv[wave * 64 + 1 * 16 + m] = a1[r];
      lv[wave * 64 + 2 * 16 + m] = a2[r];
      lv[wave * 64 + 3 * 16 + m] = a3[r];
    }
  }
  __syncthreads();
  for (int c = tid; c < 64; c += TPB) {
    float s = 0.f;
#pragma unroll
    for (int w = 0; w < 8; ++w) s += lv[w * 64 + c];
    P2[(long)blockIdx.x * 64 + c] = s;
  }
}

// Final: reduce var partials, combine all three losses
__global__ __launch_bounds__(64) void dl_final(const float* __restrict__ P2,
                                               const float* __restrict__ S,
                                               const float* __restrict__ scal,
                                               float* __restrict__ out) {
  __shared__ float red[64];
  const int tid = threadIdx.x;  // 64 threads
  float v = 0.f;
  for (int b = 0; b < NBLK; ++b) v += P2[(long)b * 64 + tid];
  v /= fmaxf(S[1024 + tid], 1.f);
  red[tid] = v;
  __syncthreads();
  for (int s = 32; s > 0; s >>= 1) {
    if (tid < s) red[tid] += red[tid + s];
    __syncthreads();
  }
  if (tid == 0) {
    float var_loss  = red[0] / 64.f;
    float dist_loss = scal[0];
    float reg_loss  = scal[1];
    out[0] = var_loss + dist_loss + 0.001f * reg_loss;
    out[1] = var_loss;
    out[2] = dist_loss;
    out[3] = reg_loss;
  }
}

extern "C" void kernel_launch(void* const* d_in, const int* in_sizes, int n_in,
                              void* d_out, int out_size, void* d_ws, size_t ws_size,
                              hipStream_t stream) {
  const float* feat = (const float*)d_in[0];
  const int*   lab  = (const int*)d_in[1];
  const int N = in_sizes[1];  // labels count = N (features = N*16)

  float* ws    = (float*)d_ws;
  float* P1    = ws;                          // 256 * 1088
  float* P2    = P1 + (long)NBLK * P1_STRIDE; // 256 * 64
  float* S     = P2 + (long)NBLK * 64;        // 1088
  float* means = S + P1_STRIDE;               // 1024
  float* scal  = means + 1024;                // 2

  dl_pass1<<<NBLK, TPB, 0, stream>>>(feat, lab, P1, N);
  dl_reduce1<<<(P1_STRIDE + 255) / 256, 256, 0, stream>>>(P1, S);
  dl_means_pairs<<<1, 256, 0, stream>>>(S, means, scal);
  dl_pass2<<<NBLK, TPB, 0, stream>>>(feat, lab, means, P2, N);
  dl_final<<<1, 64, 0, stream>>>(P2, S, scal, (float*)d_out);
}